// Model_35235911696437
// MI455X (gfx1250) — compile-verified
//
#include <hip/hip_runtime.h>

typedef __attribute__((ext_vector_type(2))) float v2f;
typedef __attribute__((ext_vector_type(8))) float v8f;

#define NGATES 30

// Gate element G[row][col] of the 16x16 real matrix for each gate type.
// Wire w <-> bit position (3-w) of the flat index (axis 0 is most significant).
// type: 0=X(p0), 1=H(p0), 2=RY on wire p0 with sin-sign p1, 3=CNOT(ctrl=p0,tgt=p1)
__device__ __forceinline__ float gate_elem(int type, int p0, int p1,
                                           float c, float s, int row, int col) {
  if (type == 3) {  // CNOT: column j maps to row j (ctrl=0) or j^tgtmask (ctrl=1)
    int bc = 3 - p0, bt = 3 - p1;
    int i0 = ((col >> bc) & 1) ? (col ^ (1 << bt)) : col;
    return (row == i0) ? 1.0f : 0.0f;
  }
  int bw = 3 - p0;
  int mask = 1 << bw;
  if ((row & ~mask) != (col & ~mask)) return 0.0f;
  int bi = (row >> bw) & 1;
  int bj = (col >> bw) & 1;
  if (type == 0) return (bi != bj) ? 1.0f : 0.0f;  // X
  if (type == 1) {                                  // H
    const float SQ2 = 0.70710678118654752440f;
    return (bi && bj) ? -SQ2 : SQ2;
  }
  // RY(sign*phi/8): [[c, -sg*s],[sg*s, c]] with c=cos(phi/16), s=sin(phi/16)
  float sg = (float)p1;
  if (bi == bj) return c;
  return bi ? (sg * s) : (-sg * s);
}

__global__ __launch_bounds__(32)
void vqe_energy_kernel(const float* __restrict__ phi_in, float* __restrict__ out) {
  __shared__ float lM[16][16];  // accumulated unitary, row-major

  const int l = threadIdx.x;
  const int n = l & 15;   // column (B/C/D) or A-row index
  const int h = l >> 4;   // lane half

  const float phi = phi_in[0];
  float s, c;
  sincosf(phi * 0.0625f, &s, &c);  // theta/2 = (phi/8)/2 = phi/16

  // Circuit as (type, p0, p1) triples, in application order.
  const int GT[NGATES]  = {0,0,3,3,1,1,3,3,2,2, 3,1,3,2,2,3,3,2,2,3, 1,3,2,2,3,3,1,1,3,3};
  const int GP0[NGATES] = {0,1,2,0,3,0,2,0,1,0, 0,3,3,1,0,2,2,1,0,3, 3,0,1,0,0,2,0,3,0,2};
  const int GP1[NGATES] = {0,0,3,2,0,0,3,1,1,-1, 3,0,1,1,-1,1,0,-1,1,1, 0,3,-1,1,1,0,0,0,2,3};

  // M = identity (each lane owns its C/D-layout elements: row r+8h, col n)
  #pragma unroll
  for (int r = 0; r < 8; ++r) {
    int row = r + 8 * h;
    lM[row][n] = (row == n) ? 1.0f : 0.0f;
  }
  __syncthreads();

  for (int g = 0; g < NGATES; ++g) {
    const int t = GT[g], q0 = GP0[g], q1 = GP1[g];

    // A fragments: A[m=n][K = 4k + r + 2h], computed from the gate formula.
    // B fragments: B[K = 4k + r + 2h][n], loaded from M in LDS.
    v2f a0, a1, a2, a3, b0, b1, b2, b3;
    a0.x = gate_elem(t, q0, q1, c, s, n, 0 + 2 * h);
    a0.y = gate_elem(t, q0, q1, c, s, n, 1 + 2 * h);
    a1.x = gate_elem(t, q0, q1, c, s, n, 4 + 2 * h);
    a1.y = gate_elem(t, q0, q1, c, s, n, 5 + 2 * h);
    a2.x = gate_elem(t, q0, q1, c, s, n, 8 + 2 * h);
    a2.y = gate_elem(t, q0, q1, c, s, n, 9 + 2 * h);
    a3.x = gate_elem(t, q0, q1, c, s, n, 12 + 2 * h);
    a3.y = gate_elem(t, q0, q1, c, s, n, 13 + 2 * h);

    b0.x = lM[0 + 2 * h][n];  b0.y = lM[1 + 2 * h][n];
    b1.x = lM[4 + 2 * h][n];  b1.y = lM[5 + 2 * h][n];
    b2.x = lM[8 + 2 * h][n];  b2.y = lM[9 + 2 * h][n];
    b3.x = lM[12 + 2 * h][n]; b3.y = lM[13 + 2 * h][n];

    // D = G x M  (four chained K=4 f32 WMMAs)
    v8f d = {};
    d = __builtin_amdgcn_wmma_f32_16x16x4_f32(false, a0, false, b0, (short)0, d, false, false);
    d = __builtin_amdgcn_wmma_f32_16x16x4_f32(false, a1, false, b1, (short)0, d, false, false);
    d = __builtin_amdgcn_wmma_f32_16x16x4_f32(false, a2, false, b2, (short)0, d, false, false);
    d = __builtin_amdgcn_wmma_f32_16x16x4_f32(false, a3, false, b3, (short)0, d, false, false);

    __syncthreads();
    #pragma unroll
    for (int r = 0; r < 8; ++r) lM[r + 8 * h][n] = d[r];
    __syncthreads();
  }

  // psi = M[:,0] (psi0 = e0). Lane 0 finishes the 14 Pauli expectations.
  if (l == 0) {
    float p[16];
    #pragma unroll
    for (int i = 0; i < 16; ++i) p[i] = lM[i][0];

    auto bitw = [](int i, int w) { return (i >> (3 - w)) & 1; };

    float E[14];
    // Z on wires 2,3,0,1
    const int zw[4] = {2, 3, 0, 1};
    for (int k = 0; k < 4; ++k) {
      float acc = 0.0f;
      for (int i = 0; i < 16; ++i)
        acc += p[i] * p[i] * (bitw(i, zw[k]) ? -1.0f : 1.0f);
      E[k] = acc;
    }
    // ZZ on (0,2),(1,3),(0,3),(1,2),(0,1),(2,3)
    const int za[6] = {0, 1, 0, 1, 0, 2};
    const int zb[6] = {2, 3, 3, 2, 1, 3};
    for (int k = 0; k < 6; ++k) {
      float acc = 0.0f;
      for (int i = 0; i < 16; ++i)
        acc += p[i] * p[i] * ((bitw(i, za[k]) ^ bitw(i, zb[k])) ? -1.0f : 1.0f);
      E[4 + k] = acc;
    }
    // 4-body with two Y's: j = i^15, coeff = -1 if the two y-wire bits match.
    // yyxx:(0,1)  xxyy:(2,3)  yxxy:(0,3)  xyyx:(1,2)
    const int ya[4] = {0, 2, 0, 1};
    const int yb[4] = {1, 3, 3, 2};
    for (int k = 0; k < 4; ++k) {
      float acc = 0.0f;
      for (int i = 0; i < 16; ++i) {
        float cf = (bitw(i, ya[k]) == bitw(i, yb[k])) ? -1.0f : 1.0f;
        acc += p[i] * cf * p[i ^ 15];
      }
      E[10 + k] = acc;
    }

    const float CF[14] = {
        -0.24274501260941383f, -0.24274501260941383f,
         0.1777135822909176f,   0.1777135822909176f,
         0.12293330449299354f,  0.12293330449299354f,
         0.1676833885560135f,   0.1676833885560135f,
         0.17059759276836806f,  0.17627661394181787f,
        -0.04475008406301996f, -0.04475008406301996f,
         0.04475008406301996f,  0.04475008406301996f};
    float energy = -0.042072551947440084f;
    for (int k = 0; k < 14; ++k) energy += CF[k] * E[k];
    out[0] = energy;
  }
}

extern "C" void kernel_launch(void* const* d_in, const int* in_sizes, int n_in,
                              void* d_out, int out_size, void* d_ws, size_t ws_size,
                              hipStream_t stream) {
  (void)in_sizes; (void)n_in; (void)d_ws; (void)ws_size; (void)out_size;
  const float* phi = (const float*)d_in[0];
  float* out = (float*)d_out;
  // One wave32: the whole problem is a latency-bound micro-kernel.
  vqe_energy_kernel<<<1, 32, 0, stream>>>(phi, out);
}